// GAT_Model_128849019283
// MI455X (gfx1250) — compile-verified
//
#include <hip/hip_runtime.h>
#include <hip/hip_bf16.h>

// ---------------------------------------------------------------------------
// GATv2 (2 layers, heads=1) + linear classifier for MI455X (gfx1250, wave32).
// All matmuls (node transforms x@W, per-edge edge_attr@We) run on the matrix
// pipe via V_WMMA_F32_16X16X4_F32 (fp32 in/out; the workload is bandwidth /
// atomic bound, so fp32 WMMA is the right precision path).
// ---------------------------------------------------------------------------

typedef __attribute__((ext_vector_type(2))) float v2f;
typedef __attribute__((ext_vector_type(8))) float v8f;

#define NODES  50000
#define NEDGE  1600000
#define EALL   1650000      // NEDGE + NODES self loops
#define DH     64
#define DE     32
#define NEG_SLOPE 0.2f

// order-preserving float->u32 key for atomic max on signed floats
__device__ __forceinline__ unsigned fkey(float f) {
  unsigned u = __float_as_uint(f);
  return (u & 0x80000000u) ? ~u : (u | 0x80000000u);
}
__device__ __forceinline__ float funkey(unsigned u) {
  return __uint_as_float((u & 0x80000000u) ? (u & 0x7FFFFFFFu) : ~u);
}

// ---------------------------------------------------------------------------
__global__ void zero_u32(unsigned* __restrict__ p, long long n) {
  long long t = blockIdx.x * (long long)blockDim.x + threadIdx.x;
  if (t < n) p[t] = 0u;
}

// deg[d] += 1 ; sums[d][k] += edge_attr[e][k]   (thread per (edge,k))
__global__ void loop_accum(const float* __restrict__ ea, const int* __restrict__ dstI,
                           float* __restrict__ sums, float* __restrict__ deg) {
  long long t = blockIdx.x * (long long)blockDim.x + threadIdx.x;
  if (t >= (long long)NEDGE * DE) return;
  int e = (int)(t >> 5);
  int k = (int)(t & 31);
  int d = dstI[e];
  if (k == 0) atomicAdd(&deg[d], 1.0f);
  atomicAdd(&sums[(size_t)d * DE + k], ea[(size_t)e * DE + k]);
}

// loop_attr = sums / max(deg,1)   (in place)
__global__ void loop_finalize(float* __restrict__ sums, const float* __restrict__ deg) {
  int t = blockIdx.x * blockDim.x + threadIdx.x;
  if (t >= NODES * DE) return;
  float dd = deg[t >> 5];
  dd = dd > 1.0f ? dd : 1.0f;
  sums[t] = sums[t] / dd;
}

// ---------------------------------------------------------------------------
// OUT[M,64] = A[M,K] @ W[K,64] + bias[64]  via V_WMMA_F32_16X16X4_F32.
// One wave computes a 16x64 output tile. K in {128, 64}, M multiple of 16.
// A frag (16x4): lane(r=lane&15, hi=lane>>4) holds A[r][k0+2hi], A[r][k0+2hi+1]
// B frag (4x16): lane holds W[k0+2hi][col], W[k0+2hi+1][col], col = lane&15
// C/D (16x16):   vgpr v, lane -> row v+8hi, col lane&15
// ---------------------------------------------------------------------------
__global__ void gemm_wmma(const float* __restrict__ A, const float* __restrict__ W,
                          const float* __restrict__ bias, float* __restrict__ OUT,
                          int M, int K) {
  int wave = (blockIdx.x * blockDim.x + threadIdx.x) >> 5;
  int lane = threadIdx.x & 31;
  int row0 = wave * 16;
  if (row0 >= M) return;                 // wave-uniform: EXEC stays all-ones
  int r  = lane & 15;
  int hi = lane >> 4;

  v8f acc[4] = {};
  const float* arow = A + (size_t)(row0 + r) * K;
  for (int k0 = 0; k0 < K; k0 += 4) {
    v2f a;
    a.x = arow[k0 + 2 * hi];
    a.y = arow[k0 + 2 * hi + 1];
#pragma unroll
    for (int nt = 0; nt < 4; ++nt) {
      const float* wp = W + (size_t)(k0 + 2 * hi) * DH + nt * 16 + r;
      v2f b;
      b.x = wp[0];
      b.y = wp[DH];
      acc[nt] = __builtin_amdgcn_wmma_f32_16x16x4_f32(
          false, a, false, b, (short)0, acc[nt], false, false);
    }
  }
#pragma unroll
  for (int nt = 0; nt < 4; ++nt) {
    int col = nt * 16 + r;
    float bv = bias[col];
#pragma unroll
    for (int v = 0; v < 8; ++v) {
      int m = v + 8 * hi;
      OUT[(size_t)(row0 + m) * DH + col] = acc[nt][v] + bv;
    }
  }
}

// ---------------------------------------------------------------------------
// Fused edge score: one wave handles 16 edges.
//   eW  = ea_tile(16x32) @ We(32x64)                (WMMA, 8 k-steps x 4 n-tiles)
//   z   = eW + xl[src] + xr[dst]                    (gather onto C layout)
//   s   = sum_col att[col] * leaky(z)               (shfl reduction per row)
//   score[e] = s ; atomicMax(mkey[dst], key(s))
// Edges e >= NEDGE are self loops: src=dst=e-NEDGE, attr = loop_attr row.
// ---------------------------------------------------------------------------
__global__ void edge_score_wmma(const float* __restrict__ ea,
                                const float* __restrict__ loopA,
                                const int* __restrict__ srcI,
                                const int* __restrict__ dstI,
                                const float* __restrict__ We,
                                const float* __restrict__ att,
                                const float* __restrict__ xl,
                                const float* __restrict__ xr,
                                float* __restrict__ score,
                                unsigned* __restrict__ mkey) {
  int wave = (blockIdx.x * blockDim.x + threadIdx.x) >> 5;
  int lane = threadIdx.x & 31;
  int e0 = wave * 16;
  if (e0 >= EALL) return;                // wave-uniform
  int r  = lane & 15;
  int hi = lane >> 4;

  // A row this lane feeds (edge e0+r)
  int er = e0 + r;
  const float* arow = (er < NEDGE) ? (ea + (size_t)er * DE)
                                   : (loopA + (size_t)(er - NEDGE) * DE);

  v8f acc[4] = {};
#pragma unroll
  for (int k0 = 0; k0 < DE; k0 += 4) {
    v2f a;
    a.x = arow[k0 + 2 * hi];
    a.y = arow[k0 + 2 * hi + 1];
#pragma unroll
    for (int nt = 0; nt < 4; ++nt) {
      const float* wp = We + (size_t)(k0 + 2 * hi) * DH + nt * 16 + r;
      v2f b;
      b.x = wp[0];
      b.y = wp[DH];
      acc[nt] = __builtin_amdgcn_wmma_f32_16x16x4_f32(
          false, a, false, b, (short)0, acc[nt], false, false);
    }
  }

  // src/dst for the 8 rows this lane holds in the C/D layout
  int sA[8], dA[8];
#pragma unroll
  for (int v = 0; v < 8; ++v) {
    int em = e0 + v + 8 * hi;
    int node = em - NEDGE;
    sA[v] = (em < NEDGE) ? srcI[em] : node;
    dA[v] = (em < NEDGE) ? dstI[em] : node;
  }

  float p[8];
#pragma unroll
  for (int v = 0; v < 8; ++v) p[v] = 0.0f;

#pragma unroll
  for (int nt = 0; nt < 4; ++nt) {
    int col = nt * 16 + r;
    float ac = att[col];
#pragma unroll
    for (int v = 0; v < 8; ++v) {
      float z = acc[nt][v]
              + xl[(size_t)sA[v] * DH + col]
              + xr[(size_t)dA[v] * DH + col];
      z = (z > 0.0f) ? z : NEG_SLOPE * z;
      p[v] += ac * z;
    }
  }

  // reduce each p[v] over the 16 lanes of this half-wave (offsets < 16)
#pragma unroll
  for (int off = 1; off < 16; off <<= 1) {
#pragma unroll
    for (int v = 0; v < 8; ++v) p[v] += __shfl_xor(p[v], off, 32);
  }

  if (r == 0) {
#pragma unroll
    for (int v = 0; v < 8; ++v) {
      int em = e0 + v + 8 * hi;
      float s = p[v];
      score[em] = s;
      atomicMax(&mkey[dA[v]], fkey(s));
    }
  }
}

// ex = exp(score - m[dst]) ; denom[dst] += ex
__global__ void edge_exp(const float* __restrict__ score, const int* __restrict__ dstI,
                         const unsigned* __restrict__ mkey,
                         float* __restrict__ ex, float* __restrict__ denom) {
  int e = blockIdx.x * blockDim.x + threadIdx.x;
  if (e >= EALL) return;
  int d = (e < NEDGE) ? dstI[e] : (e - NEDGE);
  float m = funkey(mkey[d]);
  float v = __expf(score[e] - m);
  ex[e] = v;
  atomicAdd(&denom[d], v);
}

// agg[dst][c] += (ex/denom[dst]) * xl[src][c]   (thread per (edge,c))
__global__ void edge_aggregate(const float* __restrict__ ex, const float* __restrict__ denom,
                               const int* __restrict__ srcI, const int* __restrict__ dstI,
                               const float* __restrict__ xl, float* __restrict__ agg) {
  long long t = blockIdx.x * (long long)blockDim.x + threadIdx.x;
  if (t >= (long long)EALL * DH) return;
  int e = (int)(t >> 6);
  int c = (int)(t & 63);
  int node = e - NEDGE;
  int s = (e < NEDGE) ? srcI[e] : node;
  int d = (e < NEDGE) ? dstI[e] : node;
  float alpha = ex[e] / denom[d];
  atomicAdd(&agg[(size_t)d * DH + c], alpha * xl[(size_t)s * DH + c]);
}

// h = relu(agg + bias)
__global__ void bias_relu(const float* __restrict__ agg, const float* __restrict__ bias,
                          float* __restrict__ h) {
  int t = blockIdx.x * blockDim.x + threadIdx.x;
  if (t >= NODES * DH) return;
  float v = agg[t] + bias[t & 63];
  h[t] = v > 0.0f ? v : 0.0f;
}

// out[i][0..1] = h[i] @ Wc + bc
__global__ void classifier(const float* __restrict__ h, const float* __restrict__ Wc,
                           const float* __restrict__ bc, float* __restrict__ out) {
  int i = blockIdx.x * blockDim.x + threadIdx.x;
  if (i >= NODES) return;
  const float* hr = h + (size_t)i * DH;
  float a0 = bc[0], a1 = bc[1];
#pragma unroll 8
  for (int k = 0; k < DH; ++k) {
    float v = hr[k];
    a0 += v * Wc[k * 2 + 0];
    a1 += v * Wc[k * 2 + 1];
  }
  out[(size_t)i * 2 + 0] = a0;
  out[(size_t)i * 2 + 1] = a1;
}

// ---------------------------------------------------------------------------
extern "C" void kernel_launch(void* const* d_in, const int* in_sizes, int n_in,
                              void* d_out, int out_size, void* d_ws, size_t ws_size,
                              hipStream_t stream) {
  (void)in_sizes; (void)n_in; (void)out_size; (void)ws_size;

  const float* x    = (const float*)d_in[0];
  const int*   eidx = (const int*)d_in[1];
  const float* ea   = (const float*)d_in[2];
  const float* Wl1  = (const float*)d_in[3];
  const float* bl1  = (const float*)d_in[4];
  const float* Wr1  = (const float*)d_in[5];
  const float* br1  = (const float*)d_in[6];
  const float* We1  = (const float*)d_in[7];
  const float* att1 = (const float*)d_in[8];
  const float* bias1= (const float*)d_in[9];
  const float* Wl2  = (const float*)d_in[10];
  const float* bl2  = (const float*)d_in[11];
  const float* Wr2  = (const float*)d_in[12];
  const float* br2  = (const float*)d_in[13];
  const float* We2  = (const float*)d_in[14];
  const float* att2 = (const float*)d_in[15];
  const float* bias2= (const float*)d_in[16];
  const float* Wc   = (const float*)d_in[17];
  const float* bc   = (const float*)d_in[18];
  float* out = (float*)d_out;

  const int* srcI = eidx;          // edge_index[0]
  const int* dstI = eidx + NEDGE;  // edge_index[1]

  // workspace carve-up (256B aligned)
  char* base = (char*)d_ws;
  size_t off = 0;
  auto carve = [&](size_t bytes) -> void* {
    off = (off + 255) & ~(size_t)255;
    void* p = base + off;
    off += bytes;
    return p;
  };
  float*    deg   = (float*)carve((size_t)NODES * 4);
  float*    loopA = (float*)carve((size_t)NODES * DE * 4);
  float*    xl    = (float*)carve((size_t)NODES * DH * 4);
  float*    xr    = (float*)carve((size_t)NODES * DH * 4);
  float*    h     = (float*)carve((size_t)NODES * DH * 4);
  float*    agg   = (float*)carve((size_t)NODES * DH * 4);
  float*    score = (float*)carve((size_t)EALL * 4);
  float*    ex    = (float*)carve((size_t)EALL * 4);
  unsigned* mkey  = (unsigned*)carve((size_t)NODES * 4);
  float*    denom = (float*)carve((size_t)NODES * 4);

  const int TB = 256;
  auto blocksLL = [](long long n, int tb) { return (unsigned)((n + tb - 1) / tb); };

  // --- self-loop attr (shared by both layers) ---
  zero_u32<<<blocksLL((long long)NODES, TB), TB, 0, stream>>>((unsigned*)deg, NODES);
  zero_u32<<<blocksLL((long long)NODES * DE, TB), TB, 0, stream>>>((unsigned*)loopA, (long long)NODES * DE);
  loop_accum<<<blocksLL((long long)NEDGE * DE, TB), TB, 0, stream>>>(ea, dstI, loopA, deg);
  loop_finalize<<<blocksLL((long long)NODES * DE, TB), TB, 0, stream>>>(loopA, deg);

  const unsigned gemmBlk = blocksLL((long long)(NODES / 16) * 32, TB);
  const unsigned edgeBlk = blocksLL((long long)(EALL / 16) * 32, TB);
  const unsigned eaBlk   = blocksLL((long long)EALL, TB);
  const unsigned aggBlk  = blocksLL((long long)EALL * DH, TB);
  const unsigned nhBlk   = blocksLL((long long)NODES * DH, TB);

  // ---------------- layer 1 ----------------
  gemm_wmma<<<gemmBlk, TB, 0, stream>>>(x, Wl1, bl1, xl, NODES, 128);
  gemm_wmma<<<gemmBlk, TB, 0, stream>>>(x, Wr1, br1, xr, NODES, 128);
  zero_u32<<<blocksLL((long long)NODES, TB), TB, 0, stream>>>(mkey, NODES);
  zero_u32<<<blocksLL((long long)NODES, TB), TB, 0, stream>>>((unsigned*)denom, NODES);
  zero_u32<<<nhBlk, TB, 0, stream>>>((unsigned*)agg, (long long)NODES * DH);
  edge_score_wmma<<<edgeBlk, TB, 0, stream>>>(ea, loopA, srcI, dstI, We1, att1, xl, xr, score, mkey);
  edge_exp<<<eaBlk, TB, 0, stream>>>(score, dstI, mkey, ex, denom);
  edge_aggregate<<<aggBlk, TB, 0, stream>>>(ex, denom, srcI, dstI, xl, agg);
  bias_relu<<<nhBlk, TB, 0, stream>>>(agg, bias1, h);

  // ---------------- layer 2 ----------------
  gemm_wmma<<<gemmBlk, TB, 0, stream>>>(h, Wl2, bl2, xl, NODES, 64);
  gemm_wmma<<<gemmBlk, TB, 0, stream>>>(h, Wr2, br2, xr, NODES, 64);
  zero_u32<<<blocksLL((long long)NODES, TB), TB, 0, stream>>>(mkey, NODES);
  zero_u32<<<blocksLL((long long)NODES, TB), TB, 0, stream>>>((unsigned*)denom, NODES);
  zero_u32<<<nhBlk, TB, 0, stream>>>((unsigned*)agg, (long long)NODES * DH);
  edge_score_wmma<<<edgeBlk, TB, 0, stream>>>(ea, loopA, srcI, dstI, We2, att2, xl, xr, score, mkey);
  edge_exp<<<eaBlk, TB, 0, stream>>>(score, dstI, mkey, ex, denom);
  edge_aggregate<<<aggBlk, TB, 0, stream>>>(ex, denom, srcI, dstI, xl, agg);
  bias_relu<<<nhBlk, TB, 0, stream>>>(agg, bias2, h);

  // ---------------- classifier ----------------
  classifier<<<blocksLL((long long)NODES, TB), TB, 0, stream>>>(h, Wc, bc, out);
}